// GraphLayer_70703751627242
// MI455X (gfx1250) — compile-verified
//
#include <hip/hip_runtime.h>

// MI455X / gfx1250, wave32. SpMM scatter (COO) + fused bias/relu/mask epilogue.
// Memory-bound: accumulator (76.8MB) is L2-resident (192MB L2), W (115KB)
// staged in LDS (320KB/WGP). Per-nnz scaling done bit-exactly with
// V_WMMA_F32_16X16X4_F32 via D = diag(v[0:16]) x gathered-W(16x96).
// All cross-lane (bpermute) work hoisted out of the 6-tile column loop.

typedef __attribute__((ext_vector_type(2))) float v2f;
typedef __attribute__((ext_vector_type(8))) float v8f;

#define WAVE_SZ 32
#define DOUT 96

__global__ void zero_out_kernel(float4* __restrict__ out4, int n4) {
  int i = blockIdx.x * blockDim.x + threadIdx.x;
  int stride = gridDim.x * blockDim.x;
  float4 z;
  z.x = 0.0f; z.y = 0.0f; z.z = 0.0f; z.w = 0.0f;
  for (; i < n4; i += stride) out4[i] = z;
}

// One wave per group of 16 nonzeros.
// contrib[k, 0:96] = v_k * W[c_k, 0:96], computed as six 16x16 WMMA tiles:
//   D(16x16) = A(16x16 = diag(v)) * B(16x16 = W rows gathered for cols t*16..)
// then atomicAdd D into out[r_k, :] (16 consecutive floats per row per tile).
__global__ void spmm_wmma_scatter(const float* __restrict__ xv,
                                  const int* __restrict__ xr,
                                  const int* __restrict__ xc,
                                  const float* __restrict__ W,
                                  float* __restrict__ out,
                                  int nnz, int wElems) {
  extern __shared__ float sW[];
  // Stage W (D_IN x 96 fp32, 115200 B) into LDS once per block.
  for (int i = threadIdx.x; i < wElems; i += blockDim.x) sW[i] = W[i];
  __syncthreads();

  const int lane = threadIdx.x & (WAVE_SZ - 1);
  const int wavesPerBlock = blockDim.x / WAVE_SZ;
  const int waveId = blockIdx.x * wavesPerBlock + (threadIdx.x / WAVE_SZ);
  const int totalWaves = gridDim.x * wavesPerBlock;
  const int groups = (nnz + 15) / 16;

  const int halfSel = lane >> 4;   // 0: lanes 0-15, 1: lanes 16-31
  const int nIdx = lane & 15;      // N (column-in-tile) index; also M for A

  for (int g = waveId; g < groups; g += totalWaves) {
    const int base = g * 16;
    float v = 0.0f;
    int rOff = 0, cOff = 0;        // r*96, c*96 (pre-scaled before shuffle)
    if (lane < 16) {
      int idx = base + lane;
      if (idx < nnz) {
        v = xv[idx];
        rOff = xr[idx] * DOUT;
        cOff = xc[idx] * DOUT;
      }
    }
    // v for logical row m = nIdx (replicated to both lane halves)
    const float vm = __shfl(v, nIdx, WAVE_SZ);

    // ---- tile-invariant cross-lane precompute (hoisted bpermutes) ----
    // A fragments: A[m][K] = (K==m) ? v_m : 0 ; K = kc*4 + j + 2*halfSel
    float aFrag[8];
    // B gather bases: W row offset for B-register slot (kc,j), + column lane
    int bOff[8];
#pragma unroll
    for (int kc = 0; kc < 4; ++kc) {
#pragma unroll
      for (int j = 0; j < 2; ++j) {
        const int K = kc * 4 + j + 2 * halfSel;        // global K in 0..15
        aFrag[kc * 2 + j] = (K == nIdx) ? vm : 0.0f;   // diag(v)
        bOff[kc * 2 + j] = __shfl(cOff, K, WAVE_SZ) + nIdx;
      }
    }
    // Scatter bases: D VGPR d -> row m = d + 8*halfSel, col = nIdx
    int sOff[8];
#pragma unroll
    for (int d = 0; d < 8; ++d)
      sOff[d] = __shfl(rOff, d + 8 * halfSel, WAVE_SZ) + nIdx;

    // ---- 6 column tiles of 16 ----
    for (int t = 0; t < 6; ++t) {
      const int col0 = t * 16;
      v8f acc = {};
#pragma unroll
      for (int kc = 0; kc < 4; ++kc) {
        v2f a, b;
#pragma unroll
        for (int j = 0; j < 2; ++j) {
          a[j] = aFrag[kc * 2 + j];
          b[j] = sW[bOff[kc * 2 + j] + col0];
        }
        acc = __builtin_amdgcn_wmma_f32_16x16x4_f32(
            false, a, false, b, (short)0, acc, false, false);
      }
#pragma unroll
      for (int d = 0; d < 8; ++d)
        atomicAdd(&out[sOff[d] + col0], acc[d]);
    }
  }
}

// out[i,j] = mask[i] * relu(out[i,j] + b[j]), in place, float4 vectorized.
// 96 % 4 == 0 so each float4 stays within one row and b offset is 16B aligned.
__global__ void epilogue_kernel(float* __restrict__ out,
                                const float* __restrict__ mask,
                                const float* __restrict__ bias,
                                int nTotal) {
  int i = blockIdx.x * blockDim.x + threadIdx.x;
  int stride = gridDim.x * blockDim.x;
  const int n4 = nTotal / 4;
  float4* out4 = reinterpret_cast<float4*>(out);
  for (; i < n4; i += stride) {
    const int e = i * 4;
    const int row = e / DOUT;
    const int j = e - row * DOUT;
    float4 h = out4[i];
    const float4 bb = *reinterpret_cast<const float4*>(bias + j);
    const float mk = mask[row];
    h.x = fmaxf(h.x + bb.x, 0.0f) * mk;
    h.y = fmaxf(h.y + bb.y, 0.0f) * mk;
    h.z = fmaxf(h.z + bb.z, 0.0f) * mk;
    h.w = fmaxf(h.w + bb.w, 0.0f) * mk;
    out4[i] = h;
  }
}

extern "C" void kernel_launch(void* const* d_in, const int* in_sizes, int n_in,
                              void* d_out, int out_size, void* d_ws, size_t ws_size,
                              hipStream_t stream) {
  // setup_inputs order: x_values, mask, W, b, x_row, x_col
  const float* xv   = (const float*)d_in[0];
  const float* mask = (const float*)d_in[1];
  const float* W    = (const float*)d_in[2];
  const float* bias = (const float*)d_in[3];
  const int*   xr   = (const int*)d_in[4];
  const int*   xc   = (const int*)d_in[5];
  float* out = (float*)d_out;

  const int nnz = in_sizes[0];
  const int wElems = in_sizes[2];      // D_IN * 96
  const int total = out_size;          // N * 96

  // 1) zero the accumulator (d_out)
  const int n4 = total / 4;
  const int tz = 256;
  const int gz = (n4 + tz - 1) / tz;
  zero_out_kernel<<<gz, tz, 0, stream>>>((float4*)out, n4);

  // 2) WMMA scatter-accumulate
  const int groups = (nnz + 15) / 16;
  const int threads = 256;             // 8 waves per block
  int blocks = (groups + 7) / 8;
  if (blocks > 1024) blocks = 1024;    // cap so LDS W-staging cost stays small
  const size_t shmem = (size_t)wElems * sizeof(float);  // 115,200 B < 320 KB
  (void)hipFuncSetAttribute((const void*)spmm_wmma_scatter,
                            hipFuncAttributeMaxDynamicSharedMemorySize,
                            (int)shmem);
  spmm_wmma_scatter<<<blocks, threads, shmem, stream>>>(xv, xr, xc, W, out,
                                                        nnz, wElems);

  // 3) fused bias + relu + mask, in place on d_out
  epilogue_kernel<<<gz, tz, 0, stream>>>(out, mask, bias, total);
}